// MLP_50551765074153
// MI455X (gfx1250) — compile-verified
//
#include <hip/hip_runtime.h>
#include <math.h>

typedef __attribute__((ext_vector_type(2))) float v2f;
typedef __attribute__((ext_vector_type(8))) float v8f;

#define N_FEAT 100
#define HID 16
#define NC 2

// ---------------- small init ----------------
__global__ void misc_init_kernel(unsigned* __restrict__ keys, int* __restrict__ cursor) {
  keys[0] = 0xFFFFFFFFu;  // min identity
  keys[1] = 0u;           // max identity
  cursor[0] = 0;
}

// ---------------- degree / normalization ----------------
__global__ void deg_init_kernel(int* __restrict__ deg, int n) {
  int i = blockIdx.x * blockDim.x + threadIdx.x;
  if (i < n) deg[i] = 1;  // self loop
}

__global__ void deg_accum_kernel(const int* __restrict__ dst, int* __restrict__ deg, int e) {
  int i = blockIdx.x * blockDim.x + threadIdx.x;
  if (i < e) atomicAdd(&deg[dst[i]], 1);
}

__global__ void dinv_kernel(const int* __restrict__ deg, float* __restrict__ dinv, int n) {
  int i = blockIdx.x * blockDim.x + threadIdx.x;
  if (i < n) dinv[i] = rsqrtf((float)deg[i]);
}

// ---------------- CSR-by-destination build (segments need not be ordered) ----------------
__global__ void csr_alloc_kernel(const int* __restrict__ deg, int* __restrict__ rowstart,
                                 int* __restrict__ cur, int* __restrict__ cursor, int n) {
  int i = blockIdx.x * blockDim.x + threadIdx.x;
  if (i < n) {
    cur[i] = 0;
    rowstart[i] = atomicAdd(cursor, deg[i] - 1);  // deg includes self loop; edges only
  }
}

__global__ void csr_fill_kernel(const int* __restrict__ src, const int* __restrict__ dst,
                                const int* __restrict__ rowstart, int* __restrict__ cur,
                                int* __restrict__ csr_src, int e) {
  int i = blockIdx.x * blockDim.x + threadIdx.x;
  if (i < e) {
    int d = dst[i];
    int pos = rowstart[d] + atomicAdd(&cur[d], 1);
    csr_src[pos] = src[i];
  }
}

// ---------------- GEMM: h_lin = x @ W1  via V_WMMA_F32_16X16X4_F32 ----------------
// One wave computes a 16x16 output tile. K = 100 = 25 chunks of 4.
// A 16x4 f32: lane holds row M=lane&15, VGPR0/1 = K = 2*half+{0,1}.
// B 4x16 f32: lane holds col N=lane&15, VGPR0/1 = K = 2*half+{0,1}.
// D: VGPR j -> row j + 8*half, col = lane&15.
__global__ __launch_bounds__(256) void gemm1_wmma_kernel(
    const float* __restrict__ x, const float* __restrict__ W1,
    float* __restrict__ h, int nfull_rows)
{
  const int wave = threadIdx.x >> 5;
  const int lane = threadIdx.x & 31;
  const int tile = blockIdx.x * 8 + wave;
  const int row0 = tile * 16;
  if (row0 >= nfull_rows) return;          // wave-uniform: EXEC all-ones for WMMA
  const int half = lane >> 4;
  const int id   = lane & 15;
  const float* xrow = x + (size_t)(row0 + id) * N_FEAT;
  v8f acc = {};
#pragma unroll
  for (int c = 0; c < 25; ++c) {
    const int k = 4 * c + 2 * half;
    v2f a = *(const v2f*)(xrow + k);       // 8B-aligned: k even, row stride 400B
    v2f b;
    b.x = W1[(k    ) * HID + id];
    b.y = W1[(k + 1) * HID + id];
    acc = __builtin_amdgcn_wmma_f32_16x16x4_f32(
        /*neg_a=*/false, a, /*neg_b=*/false, b,
        /*c_mod=*/(short)0, acc, /*reuse_a=*/false, /*reuse_b=*/false);
  }
  float* hrow = h + (size_t)(row0 + 8 * half) * HID + id;
#pragma unroll
  for (int j = 0; j < 8; ++j) hrow[(size_t)j * HID] = acc[j];
}

// scalar tail for rows not covered by full 16-row tiles
__global__ void gemm1_tail_kernel(const float* __restrict__ x, const float* __restrict__ W1,
                                  float* __restrict__ h, int row_start, int nrows)
{
  int idx = blockIdx.x * blockDim.x + threadIdx.x;
  int row = row_start + idx / HID;
  int col = idx % HID;
  if (row >= nrows) return;
  float acc = 0.f;
  for (int k = 0; k < N_FEAT; ++k) acc += x[(size_t)row * N_FEAT + k] * W1[k * HID + col];
  h[(size_t)row * HID + col] = acc;
}

// ---------------- layer-1 aggregation: gather over CSR, no float atomics ----------------
// 16 lanes per node; csr_src/dinv loads are lane-uniform broadcasts, h_lin row gather
// is one coalesced 64B request per edge (L2-resident).
__global__ void gather1_kernel(const float* __restrict__ hlin, const float* __restrict__ dinv,
                               const int* __restrict__ rowstart, const int* __restrict__ deg,
                               const int* __restrict__ csr_src, float* __restrict__ hagg, int n)
{
  int t = blockIdx.x * blockDim.x + threadIdx.x;
  int node = t >> 4;
  if (node >= n) return;
  int j = t & 15;
  float dn = dinv[node];
  float acc = dn * dn * hlin[(size_t)node * HID + j];   // self loop
  int beg = rowstart[node];
  int cnt = deg[node] - 1;
  for (int k = 0; k < cnt; ++k) {
    int s = csr_src[beg + k];
    acc += dn * dinv[s] * hlin[(size_t)s * HID + j];
  }
  hagg[(size_t)node * HID + j] = acc;
}

// ---------------- h2_lin = relu(h_agg + b1) @ W2 ----------------
__global__ void lin2_kernel(const float* __restrict__ hagg, const float* __restrict__ b1,
                            const float* __restrict__ W2, float* __restrict__ h2, int n) {
  int i = blockIdx.x * blockDim.x + threadIdx.x;
  if (i >= n) return;
  const float* row = hagg + (size_t)i * HID;
  float o0 = 0.f, o1 = 0.f;
#pragma unroll
  for (int j = 0; j < HID; ++j) {
    float v = fmaxf(row[j] + b1[j], 0.f);
    o0 = fmaf(v, W2[j * NC + 0], o0);
    o1 = fmaf(v, W2[j * NC + 1], o1);
  }
  h2[(size_t)i * NC + 0] = o0;
  h2[(size_t)i * NC + 1] = o1;
}

// ---------------- layer-2 aggregation: gather over same CSR ----------------
__global__ void gather2_kernel(const float* __restrict__ h2lin, const float* __restrict__ dinv,
                               const int* __restrict__ rowstart, const int* __restrict__ deg,
                               const int* __restrict__ csr_src, float* __restrict__ h2agg, int n)
{
  int node = blockIdx.x * blockDim.x + threadIdx.x;
  if (node >= n) return;
  float dn = dinv[node];
  v2f self = *(const v2f*)(h2lin + (size_t)node * NC);
  float a0 = dn * dn * self.x;
  float a1 = dn * dn * self.y;
  int beg = rowstart[node];
  int cnt = deg[node] - 1;
  for (int k = 0; k < cnt; ++k) {
    int s = csr_src[beg + k];
    float w = dn * dinv[s];
    v2f v = *(const v2f*)(h2lin + (size_t)s * NC);
    a0 = fmaf(w, v.x, a0);
    a1 = fmaf(w, v.y, a1);
  }
  h2agg[(size_t)node * NC + 0] = a0;
  h2agg[(size_t)node * NC + 1] = a1;
}

// ---------------- head: log_softmax stride rows + global min/max ----------------
__device__ __forceinline__ unsigned f2key(float f) {
  unsigned u = __float_as_uint(f);
  return (u & 0x80000000u) ? ~u : (u | 0x80000000u);
}
__device__ __forceinline__ float key2f(unsigned k) {
  unsigned u = (k & 0x80000000u) ? (k & 0x7FFFFFFFu) : ~k;
  return __uint_as_float(u);
}

__global__ __launch_bounds__(256) void head_pre_kernel(
    const float* __restrict__ h2agg, const float* __restrict__ b2,
    const float* __restrict__ tE, const float* __restrict__ cE, const float* __restrict__ pth,
    float* __restrict__ bp, unsigned* __restrict__ keys, int m, int stride)
{
  __shared__ unsigned smin[256], smax[256];
  int i = blockIdx.x * blockDim.x + threadIdx.x;
  unsigned kmin = 0xFFFFFFFFu, kmax = 0u;
  if (i < m) {
    int n = i * stride;
    float a = h2agg[(size_t)n * NC + 0] + b2[0];
    float b = h2agg[(size_t)n * NC + 1] + b2[1];
    float mx = fmaxf(a, b);
    float ls = mx + logf(expf(a - mx) + expf(b - mx));
    float g0 = a - ls, g1 = b - ls;
    float r0 = tE[i], r1 = cE[i], r2 = pth[i];
    bp[(size_t)i * 5 + 0] = r0;
    bp[(size_t)i * 5 + 1] = r1;
    bp[(size_t)i * 5 + 2] = r2;
    bp[(size_t)i * 5 + 3] = g0;
    bp[(size_t)i * 5 + 4] = g1;
    float lmin = fminf(fminf(fminf(r0, r1), fminf(r2, g0)), g1);
    float lmax = fmaxf(fmaxf(fmaxf(r0, r1), fmaxf(r2, g0)), g1);
    kmin = f2key(lmin);
    kmax = f2key(lmax);
  }
  smin[threadIdx.x] = kmin;
  smax[threadIdx.x] = kmax;
  __syncthreads();
  for (int ofs = 128; ofs > 0; ofs >>= 1) {
    if ((int)threadIdx.x < ofs) {
      unsigned a0 = smin[threadIdx.x], b0 = smin[threadIdx.x + ofs];
      smin[threadIdx.x] = a0 < b0 ? a0 : b0;
      unsigned a1 = smax[threadIdx.x], b1 = smax[threadIdx.x + ofs];
      smax[threadIdx.x] = a1 > b1 ? a1 : b1;
    }
    __syncthreads();
  }
  if (threadIdx.x == 0) {
    atomicMin(&keys[0], smin[0]);
    atomicMax(&keys[1], smax[0]);
  }
}

// ---------------- MLP head: 5 -> 80 -> 10 -> 1, weights staged in LDS ----------------
__global__ __launch_bounds__(256) void mlp_kernel(
    const float* __restrict__ bp, const unsigned* __restrict__ keys,
    const float* __restrict__ W1, const float* __restrict__ b1,
    const float* __restrict__ W2, const float* __restrict__ b2,
    const float* __restrict__ W3, const float* __restrict__ b3,
    float* __restrict__ out, int m)
{
  __shared__ float sW1[5 * 80], sb1[80], sW2[80 * 10], sb2[10], sW3[10];
  for (int t = threadIdx.x; t < 400; t += blockDim.x) sW1[t] = W1[t];
  for (int t = threadIdx.x; t < 80;  t += blockDim.x) sb1[t] = b1[t];
  for (int t = threadIdx.x; t < 800; t += blockDim.x) sW2[t] = W2[t];
  if (threadIdx.x < 10) { sb2[threadIdx.x] = b2[threadIdx.x]; sW3[threadIdx.x] = W3[threadIdx.x]; }
  __syncthreads();
  int i = blockIdx.x * blockDim.x + threadIdx.x;
  if (i >= m) return;
  float mn = key2f(keys[0]);
  float mx = key2f(keys[1]);
  float scale = 1.0f / (mx - mn);
  float in[5];
#pragma unroll
  for (int k = 0; k < 5; ++k) in[k] = (bp[(size_t)i * 5 + k] - mn) * scale;
  float h1[80];
#pragma unroll
  for (int j = 0; j < 80; ++j) {
    float acc = sb1[j];
#pragma unroll
    for (int k = 0; k < 5; ++k) acc = fmaf(in[k], sW1[k * 80 + j], acc);
    h1[j] = fmaxf(acc, 0.f);
  }
  float h2[10];
#pragma unroll
  for (int o = 0; o < 10; ++o) {
    float acc = sb2[o];
#pragma unroll
    for (int j = 0; j < 80; ++j) acc = fmaf(h1[j], sW2[j * 10 + o], acc);
    h2[o] = fmaxf(acc, 0.f);
  }
  float acc = b3[0];
#pragma unroll
  for (int o = 0; o < 10; ++o) acc = fmaf(h2[o], sW3[o], acc);
  out[i] = 1.0f / (1.0f + expf(-acc));
}

// ---------------- launcher ----------------
extern "C" void kernel_launch(void* const* d_in, const int* in_sizes, int n_in,
                              void* d_out, int out_size, void* d_ws, size_t ws_size,
                              hipStream_t stream)
{
  (void)n_in; (void)out_size; (void)ws_size;
  const int*   edge1 = (const int*)  d_in[0];
  const float* x1    = (const float*)d_in[1];
  const float* tE    = (const float*)d_in[4];
  const float* cE    = (const float*)d_in[5];
  const float* pth   = (const float*)d_in[6];
  const float* ghW1  = (const float*)d_in[8];
  const float* ghb1  = (const float*)d_in[9];
  const float* ghW2  = (const float*)d_in[10];
  const float* ghb2  = (const float*)d_in[11];
  const float* mW1   = (const float*)d_in[16];
  const float* mb1   = (const float*)d_in[17];
  const float* mW2   = (const float*)d_in[18];
  const float* mb2   = (const float*)d_in[19];
  const float* mW3   = (const float*)d_in[20];
  const float* mb3   = (const float*)d_in[21];

  const int E = in_sizes[0] / 2;
  const int N = in_sizes[1] / N_FEAT;
  const int M = in_sizes[4];
  const int STRIDE = N / M;

  const int* src = edge1;
  const int* dst = edge1 + E;

  // workspace carve-out
  char* ws = (char*)d_ws;
  size_t off = 0;
  auto carve = [&](size_t bytes) -> void* {
    void* p = ws + off;
    off += (bytes + 255) & ~(size_t)255;
    return p;
  };
  int*      deg      = (int*)     carve((size_t)N * 4);
  float*    dinv     = (float*)   carve((size_t)N * 4);
  float*    hlin     = (float*)   carve((size_t)N * HID * 4);
  float*    hagg     = (float*)   carve((size_t)N * HID * 4);
  float*    h2lin    = (float*)   carve((size_t)N * NC * 4);
  float*    h2agg    = (float*)   carve((size_t)N * NC * 4);
  float*    bp       = (float*)   carve((size_t)M * 5 * 4);
  unsigned* keys     = (unsigned*)carve(2 * 4);
  int*      rowstart = (int*)     carve((size_t)N * 4);
  int*      cur      = (int*)     carve((size_t)N * 4);
  int*      cursor   = (int*)     carve(4);
  int*      csr_src  = (int*)     carve((size_t)E * 4);

  const int B = 256;

  // init + degrees + symmetric normalization
  misc_init_kernel<<<1, 1, 0, stream>>>(keys, cursor);
  deg_init_kernel<<<(N + B - 1) / B, B, 0, stream>>>(deg, N);
  deg_accum_kernel<<<(E + B - 1) / B, B, 0, stream>>>(dst, deg, E);
  dinv_kernel<<<(N + B - 1) / B, B, 0, stream>>>(deg, dinv, N);

  // build destination-CSR (int atomics only; segment layout order is irrelevant)
  csr_alloc_kernel<<<(N + B - 1) / B, B, 0, stream>>>(deg, rowstart, cur, cursor, N);
  csr_fill_kernel<<<(E + B - 1) / B, B, 0, stream>>>(src, dst, rowstart, cur, csr_src, E);

  // h_lin = x @ W1 (WMMA f32)
  const int nfull = N & ~15;
  const int ntiles = nfull / 16;
  if (ntiles > 0)
    gemm1_wmma_kernel<<<(ntiles + 7) / 8, B, 0, stream>>>(x1, ghW1, hlin, nfull);
  if (N != nfull) {
    int tail = (N - nfull) * HID;
    gemm1_tail_kernel<<<(tail + B - 1) / B, B, 0, stream>>>(x1, ghW1, hlin, nfull, N);
  }

  // layer-1 aggregate (gather), layer-2 linear + aggregate (gather)
  gather1_kernel<<<((long long)N * HID + B - 1) / B, B, 0, stream>>>(
      hlin, dinv, rowstart, deg, csr_src, hagg, N);
  lin2_kernel<<<(N + B - 1) / B, B, 0, stream>>>(hagg, ghb1, ghW2, h2lin, N);
  gather2_kernel<<<(N + B - 1) / B, B, 0, stream>>>(
      h2lin, dinv, rowstart, deg, csr_src, h2agg, N);

  // head: log_softmax + BP rows + global min/max, then MLP
  head_pre_kernel<<<(M + B - 1) / B, B, 0, stream>>>(h2agg, ghb2, tE, cE, pth, bp, keys, M, STRIDE);
  mlp_kernel<<<(M + B - 1) / B, B, 0, stream>>>(bp, keys, mW1, mb1, mW2, mb2, mW3, mb3,
                                                (float*)d_out, M);
}